// SourceEncoder_1125281432131
// MI455X (gfx1250) — compile-verified
//
#include <hip/hip_runtime.h>
#include <cstdint>

// ---------------------------------------------------------------------------
// Problem constants (from the reference)
// ---------------------------------------------------------------------------
namespace sc {
constexpr int SLEN = 300, PT = 20, STEP = 4, EP = 2, MAXD = 2;
constexpr int NB = 2, NS = 30, NGAL = 8, B = 8;
constexpr int N1  = (SLEN - PT) / STEP + 1;   // 71
constexpr int NPT = N1 * N1;                  // 5041
constexpr int SUB = B * NPT;                  // 40328 (== 8 * 5041 exactly)

// Flat float offsets of each tuple member in d_out (concatenated in order)
constexpr size_t OFF_PTILES = 0;
constexpr size_t OFF_LOCS   = OFF_PTILES + (size_t)SUB * NB * PT * PT; // 32,262,400
constexpr size_t OFF_GAL    = OFF_LOCS   + (size_t)SUB * MAXD * 2;     // 32,423,712
constexpr size_t OFF_FLUX   = OFF_GAL    + (size_t)SUB * MAXD * NGAL;  // 33,068,960
constexpr size_t OFF_NSRC   = OFF_FLUX   + (size_t)SUB * MAXD * NB;    // 33,230,272
constexpr size_t OFF_ISON   = OFF_NSRC   + (size_t)SUB;                // 33,270,600
} // namespace sc

typedef uint32_t v4u __attribute__((ext_vector_type(4)));
typedef uint32_t v8u __attribute__((ext_vector_type(8)));

// ---------------------------------------------------------------------------
// Kernel 1: image -> overlapping 20x20x2 tiles via the Tensor Data Mover.
// One wave per tile:
//   tensor_load_to_lds   : 3D tile (x=20, y=20 rows stride 300, z=2 planes
//                          stride 90000) from the image into a 3200B LDS slice
//   tensor_store_from_lds: 1D tile (800 contiguous elements) to the output
// TENSORcnt drains between/after. TDM ignores EXEC; descriptors are built in
// SGPRs (readfirstlane on the wave-uniform dynamic dwords).
// ---------------------------------------------------------------------------
__global__ __launch_bounds__(256) void tile_copy_tdm(const float* __restrict__ img,
                                                     float* __restrict__ out) {
  using namespace sc;
  __shared__ float lds[8 * NB * PT * PT];       // 8 waves * 800 floats = 25.6 KB
  const int wave = threadIdx.x >> 5;
  const int tile = blockIdx.x * 8 + wave;       // grid = NPT blocks -> exactly SUB tiles

  const int b  = tile / NPT;
  const int t  = tile - b * NPT;
  const int ti = t / N1;
  const int tj = t - ti * N1;

  // Byte addresses
  const uint64_t gsrc = (uint64_t)(uintptr_t)img +
                        4ull * ((uint64_t)b * (NB * SLEN * SLEN) +
                                (uint64_t)(ti * STEP) * SLEN +
                                (uint64_t)(tj * STEP));
  const uint64_t gdst = (uint64_t)(uintptr_t)out + 3200ull * (uint64_t)tile;
  const uint32_t lbase = (uint32_t)(uintptr_t)(&lds[wave * (NB * PT * PT)]);

  // Wave-uniform scalars for the SGPR descriptor groups
  const uint32_t sl  = __builtin_amdgcn_readfirstlane(lbase);
  const uint32_t sr0 = __builtin_amdgcn_readfirstlane((uint32_t)gsrc);
  const uint32_t sr1 = __builtin_amdgcn_readfirstlane((uint32_t)(gsrc >> 32));
  const uint32_t sd0 = __builtin_amdgcn_readfirstlane((uint32_t)gdst);
  const uint32_t sd1 = __builtin_amdgcn_readfirstlane((uint32_t)(gdst >> 32));

  // ---- Load descriptor: 3D tile 20 x 20 x 2 out of a 300-stride image ----
  const v4u g0L = { 1u,                                     // count=1 (valid)
                    sl,                                     // lds_addr (bytes)
                    sr0,                                    // global_addr[31:0]
                    (sr1 & 0x01FFFFFFu) | (2u << 30) };     // addr[56:32] | type=2
  const v8u g1L = { 2u << 16,                               // data_size=4B code 2
                    (uint32_t)SLEN << 16,                   // tensor_dim0 lo16 (=300)
                    (uint32_t)SLEN << 16,                   // dim0 hi16=0 | tensor_dim1 lo16 (=300)
                    (uint32_t)PT << 16,                     // dim1 hi16=0 | tile_dim0=20
                    (uint32_t)PT | ((uint32_t)NB << 16),    // tile_dim1=20 | tile_dim2=2
                    (uint32_t)SLEN,                         // tensor_dim0_stride lo32 (=300)
                    ((uint32_t)(SLEN * SLEN) & 0xFFFFu) << 16, // stride0 hi16=0 | stride1 lo16 (90000&0xFFFF)
                    (uint32_t)(SLEN * SLEN) >> 16 };        // tensor_dim1_stride hi (90000>>16 = 1)
  const v4u g2L = { (uint32_t)NB, 0u, 0u, 0u };             // tensor_dim2=2; tile_dim3=0 => unused
  const v4u g3L = { 0u, 0u, 0u, 0u };                       // unused dims (tile_dim4=0)

  asm volatile("tensor_load_to_lds %0, %1, %2, %3"
               :: "s"(g0L), "s"(g1L), "s"(g2L), "s"(g3L) : "memory");
  __builtin_amdgcn_s_wait_tensorcnt(0);                     // tile resident in LDS

  // ---- Store descriptor: 1D, 800 contiguous elements ----
  const v4u g0S = { 1u, sl, sd0, (sd1 & 0x01FFFFFFu) | (2u << 30) };
  const v8u g1S = { 2u << 16,                               // data_size=4B
                    800u << 16,                             // tensor_dim0=800 lo16
                    1u << 16,                               // dim0 hi16=0 | tensor_dim1=1
                    800u << 16,                             // dim1 hi16=0 | tile_dim0=800
                    0u,                                     // tile_dim1=0, tile_dim2=0 => 1D
                    800u,                                   // tensor_dim0_stride
                    0u, 0u };
  asm volatile("tensor_store_from_lds %0, %1"
               :: "s"(g0S), "s"(g1S) : "memory");
  __builtin_amdgcn_s_wait_tensorcnt(0);                     // drain before LDS dealloc
}

// ---------------------------------------------------------------------------
// Kernel 2: per-(b,tile) source assignment.
// Stable-argsort-on-(1-is_on) + mask  ==  keep first min(n_on, 2) on-sources
// in original index order; everything else is 0.
// ---------------------------------------------------------------------------
__global__ __launch_bounds__(256) void assign_sources(const float* __restrict__ locs,
                                                      const float* __restrict__ gal,
                                                      const float* __restrict__ flux,
                                                      float* __restrict__ out) {
  using namespace sc;
  const int s = blockIdx.x * blockDim.x + threadIdx.x;
  if (s >= SUB) return;

  const int b  = s / NPT;
  const int t  = s - b * NPT;
  const int ti = t / N1;
  const int tj = t - ti * N1;

  const float lo0 = (float)(ti * STEP) + ((float)EP - 0.5f);            // + 1.5
  const float hi0 = (float)(ti * STEP) + ((float)(PT - EP) - 0.5f);     // +17.5
  const float lo1 = (float)(tj * STEP) + ((float)EP - 0.5f);
  const float hi1 = (float)(tj * STEP) + ((float)(PT - EP) - 0.5f);
  constexpr float INV_SCALE = 1.0f / (float)(PT - 2 * EP);              // 1/16

  float tl[MAXD][2]    = {};
  float tg[MAXD][NGAL] = {};
  float tf[MAXD][NB]   = {};
  int   cnt = 0;

  const float* lb = locs + (size_t)b * NS * 2;
  for (int n = 0; n < NS; ++n) {
    const float l0 = lb[2 * n + 0] * (float)(SLEN - 1);
    const float l1 = lb[2 * n + 1] * (float)(SLEN - 1);
    const bool on = (l0 > lo0) & (l0 < hi0) & (l0 != 0.0f) &
                    (l1 > lo1) & (l1 < hi1) & (l1 != 0.0f);
    if (on) {
      if (cnt < MAXD) {
        tl[cnt][0] = fmaxf((l0 - lo0) * INV_SCALE, 0.0f);
        tl[cnt][1] = fmaxf((l1 - lo1) * INV_SCALE, 0.0f);
        const float* gp = gal  + ((size_t)b * NS + n) * NGAL;
        const float* fp = flux + ((size_t)b * NS + n) * NB;
        #pragma unroll
        for (int g = 0; g < NGAL; ++g) tg[cnt][g] = gp[g];
        #pragma unroll
        for (int c = 0; c < NB; ++c)   tf[cnt][c] = fp[c];
      }
      ++cnt;
    }
  }
  const int ns = cnt < MAXD ? cnt : MAXD;

  float* oL = out + OFF_LOCS + (size_t)s * MAXD * 2;
  float* oG = out + OFF_GAL  + (size_t)s * MAXD * NGAL;
  float* oF = out + OFF_FLUX + (size_t)s * MAXD * NB;
  #pragma unroll
  for (int k = 0; k < MAXD; ++k) {
    oL[k * 2 + 0] = tl[k][0];
    oL[k * 2 + 1] = tl[k][1];
    #pragma unroll
    for (int g = 0; g < NGAL; ++g) oG[k * NGAL + g] = tg[k][g];
    #pragma unroll
    for (int c = 0; c < NB; ++c)   oF[k * NB + c]   = tf[k][c];
  }
  out[OFF_NSRC + s] = (float)ns;
  out[OFF_ISON + (size_t)s * MAXD + 0] = (0 < ns) ? 1.0f : 0.0f;
  out[OFF_ISON + (size_t)s * MAXD + 1] = (1 < ns) ? 1.0f : 0.0f;
}

// ---------------------------------------------------------------------------
extern "C" void kernel_launch(void* const* d_in, const int* in_sizes, int n_in,
                              void* d_out, int out_size, void* d_ws, size_t ws_size,
                              hipStream_t stream) {
  (void)in_sizes; (void)n_in; (void)out_size; (void)d_ws; (void)ws_size;
  const float* images = (const float*)d_in[0];
  const float* locs   = (const float*)d_in[1];
  const float* gal    = (const float*)d_in[2];
  const float* flux   = (const float*)d_in[3];
  float* out = (float*)d_out;

  tile_copy_tdm<<<sc::NPT, 256, 0, stream>>>(images, out);               // 5041 blocks * 8 waves
  assign_sources<<<(sc::SUB + 255) / 256, 256, 0, stream>>>(locs, gal, flux, out);
}